// BatchedGNNModel_45191645888927
// MI455X (gfx1250) — compile-verified
//
#include <hip/hip_runtime.h>

// ---------------------------------------------------------------------------
// Batched GCN (B=1024, N=192, H=128) for gfx1250.
// Strategy: one workgroup per batch element; 8 wave32 per workgroup.
// All matmuls run on V_WMMA_F32_16X16X4_F32 (fp32 matrix pipe).
//   stage A: T = h @ W^T + b      (A = h tile from LDS, B = W^T cached in VGPRs)
//   stage B: h' = act(An @ T)     (A = An tile from global/L2, B = T slice in VGPRs)
// An (normalized adjacency) is batch-invariant -> computed once into d_ws.
// ---------------------------------------------------------------------------

typedef float v2f __attribute__((ext_vector_type(2)));
typedef float v8f __attribute__((ext_vector_type(8)));

#define NN    192          // nodes
#define HH    128          // hidden width
#define HS    132          // LDS row stride (pad: 132 % 64 banks = 4 -> conflict-free)
#define MT    12           // 192/16 M-tiles
#define KANT  48           // 192/4 K-steps for An aggregation

__device__ __forceinline__ v8f wmma4(v2f a, v2f b, v8f c) {
    // 8 args: (neg_a, A, neg_b, B, c_mod, C, reuse_a, reuse_b)
    return __builtin_amdgcn_wmma_f32_16x16x4_f32(false, a, false, b,
                                                 (short)0, c, false, false);
}

// ---------------- prep: An = D^-1/2 A D^-1/2 from adjacency[0] --------------
__global__ void prep_An_kernel(const float* __restrict__ adj,  // [B,192,192], batch 0 used
                               float* __restrict__ An) {       // [192,192]
    __shared__ float dinv[NN];
    const int tid = threadIdx.x;
    if (tid < NN) {
        float s = 0.f;
        for (int j = 0; j < NN; ++j) s += adj[tid * NN + j];
        dinv[tid] = (s == 0.f) ? 0.f : (1.0f / sqrtf(s));
    }
    __syncthreads();
    for (int idx = tid; idx < NN * NN; idx += blockDim.x) {
        const int i = idx / NN, j = idx % NN;
        An[idx] = adj[idx] * dinv[i] * dinv[j];
    }
}

// ---------------- stage A: T = h @ W^T + b  (Hout = 128, wave owns n-tile) --
template <int KPAD, int KREAL>
__device__ __forceinline__ void stageA_fullN(const float* __restrict__ hbuf,
                                             float* __restrict__ tbuf,
                                             const float* __restrict__ W,
                                             const float* __restrict__ bias,
                                             int wave, int lane) {
    const int n0   = 16 * wave;
    const int n    = lane & 15;
    const int kh   = 2 * (lane >> 4);       // K sub-pair selected by lane[4]
    const int col  = n0 + n;

    v2f Breg[KPAD / 4];                     // W^T column block, cached in VGPRs
#pragma unroll
    for (int ks = 0; ks < KPAD / 4; ++ks) {
        const int kk = 4 * ks + kh;
        const float w0 = (kk     < KREAL) ? W[col * KREAL + kk    ] : 0.f;
        const float w1 = (kk + 1 < KREAL) ? W[col * KREAL + kk + 1] : 0.f;
        Breg[ks] = v2f{w0, w1};
    }
    const float bv = bias[col];

    for (int mt = 0; mt < MT; ++mt) {
        const float* atile = hbuf + (16 * mt + n) * HS;   // row = m0 + (lane&15)
        v8f acc;
#pragma unroll
        for (int r = 0; r < 8; ++r) acc[r] = bv;          // bias BEFORE An multiply
#pragma unroll
        for (int ks = 0; ks < KPAD / 4; ++ks) {
            const int kk = 4 * ks + kh;
            acc = wmma4(v2f{atile[kk], atile[kk + 1]}, Breg[ks], acc);
        }
        const int rb = 16 * mt + 8 * (lane >> 4);
#pragma unroll
        for (int r = 0; r < 8; ++r) tbuf[(rb + r) * HS + col] = acc[r];
    }
}

// ---------------- stage B: h' = lrelu(An @ T)  (Hout = 128) -----------------
__device__ __forceinline__ void stageB_fullN(const float* __restrict__ tbuf,
                                             float* __restrict__ hbuf,
                                             const float* __restrict__ An,
                                             int wave, int lane, bool relu) {
    const int n0  = 16 * wave;
    const int n   = lane & 15;
    const int kh  = 2 * (lane >> 4);
    const int col = n0 + n;

    v2f Breg[KANT];                          // T[:, col] slice cached in VGPRs
#pragma unroll
    for (int ks = 0; ks < KANT; ++ks) {
        const int kk = 4 * ks + kh;
        const float* p = tbuf + kk * HS + col;
        Breg[ks] = v2f{p[0], p[HS]};
    }

    for (int mt = 0; mt < MT; ++mt) {
        const float* arow = An + (16 * mt + n) * NN;
        v8f acc = {};
#pragma unroll
        for (int ks = 0; ks < KANT; ++ks) {
            const int kk = 4 * ks + kh;
            acc = wmma4(v2f{arow[kk], arow[kk + 1]}, Breg[ks], acc);
        }
        const int rb = 16 * mt + 8 * (lane >> 4);
#pragma unroll
        for (int r = 0; r < 8; ++r) {
            float v = acc[r];
            if (relu) v = (v > 0.f) ? v : 0.2f * v;
            hbuf[(rb + r) * HS + col] = v;
        }
    }
}

// ---------------- layer 5 (Hout = 3, one n-tile padded to 16) ---------------
__device__ __forceinline__ void layer5_stageA(const float* __restrict__ hbuf,
                                              float* __restrict__ tbuf,
                                              const float* __restrict__ W5,
                                              const float* __restrict__ b5,
                                              int wave, int lane) {
    const int n  = lane & 15;
    const int kh = 2 * (lane >> 4);
    v2f Breg[HH / 4];
#pragma unroll
    for (int ks = 0; ks < HH / 4; ++ks) {
        const int kk = 4 * ks + kh;
        const float w0 = (n < 3) ? W5[n * HH + kk    ] : 0.f;
        const float w1 = (n < 3) ? W5[n * HH + kk + 1] : 0.f;
        Breg[ks] = v2f{w0, w1};
    }
    const float bv = (n < 3) ? b5[n] : 0.f;

    for (int mt = wave; mt < MT; mt += 8) {
        const float* atile = hbuf + (16 * mt + n) * HS;
        v8f acc;
#pragma unroll
        for (int r = 0; r < 8; ++r) acc[r] = bv;
#pragma unroll
        for (int ks = 0; ks < HH / 4; ++ks) {
            const int kk = 4 * ks + kh;
            acc = wmma4(v2f{atile[kk], atile[kk + 1]}, Breg[ks], acc);
        }
        const int rb = 16 * mt + 8 * (lane >> 4);
#pragma unroll
        for (int r = 0; r < 8; ++r) tbuf[(rb + r) * HS + n] = acc[r];
    }
}

__device__ __forceinline__ void layer5_stageB(const float* __restrict__ tbuf,
                                              const float* __restrict__ An,
                                              const float* __restrict__ inp_b,
                                              float* __restrict__ out_b,
                                              int wave, int lane) {
    const int n  = lane & 15;
    const int kh = 2 * (lane >> 4);
    v2f Breg[KANT];
#pragma unroll
    for (int ks = 0; ks < KANT; ++ks) {
        const int kk = 4 * ks + kh;
        const float* p = tbuf + kk * HS + n;
        Breg[ks] = v2f{p[0], p[HS]};
    }
    for (int mt = wave; mt < MT; mt += 8) {
        const float* arow = An + (16 * mt + n) * NN;
        v8f acc = {};
#pragma unroll
        for (int ks = 0; ks < KANT; ++ks) {
            const int kk = 4 * ks + kh;
            acc = wmma4(v2f{arow[kk], arow[kk + 1]}, Breg[ks], acc);
        }
        const int rb = 16 * mt + 8 * (lane >> 4);
        if (n < 3) {
#pragma unroll
            for (int r = 0; r < 8; ++r) {
                const int row = rb + r;
                const bool cl = (row == 0) | (row == 63) | (row == 127) | (row == 191);
                out_b[row * 3 + n] = cl ? inp_b[row * 3 + n] : acc[r];
            }
        }
    }
}

// ---------------- main kernel: one workgroup per batch element --------------
__global__ void __launch_bounds__(256)
gcn_main_kernel(const float* __restrict__ x,       // [B,192,6]
                const float* __restrict__ inputs,  // [B,192,3]
                const float* __restrict__ An,      // [192,192] (ws)
                const float* __restrict__ W1, const float* __restrict__ b1,
                const float* __restrict__ W2, const float* __restrict__ b2,
                const float* __restrict__ W3, const float* __restrict__ b3,
                const float* __restrict__ W4, const float* __restrict__ b4,
                const float* __restrict__ W5, const float* __restrict__ b5,
                float* __restrict__ out) {         // [B,192,3]
    extern __shared__ float smem[];
    float* hbuf = smem;                // [192][HS] activations
    float* tbuf = smem + NN * HS;      // [192][HS] h @ W^T + b

    const int b    = blockIdx.x;
    const int tid  = threadIdx.x;
    const int lane = tid & 31;         // wave32
    const int wave = tid >> 5;         // 0..7

    const float* x_b   = x      + (size_t)b * NN * 6;
    const float* inp_b = inputs + (size_t)b * NN * 3;
    float*       out_b = out    + (size_t)b * NN * 3;

    // --- load layer-1 input, apply boundary clamp, pad K to 8 with zeros ---
    for (int row = tid; row < NN; row += blockDim.x) {
        const bool cl = (row == 0) | (row == 63) | (row == 127) | (row == 191);
        float* hr = hbuf + row * HS;
#pragma unroll
        for (int f = 0; f < 6; ++f) {
            float v = x_b[row * 6 + f];
            if (cl && f < 3) v = inp_b[row * 3 + f];
            hr[f] = v;
        }
        hr[6] = 0.f; hr[7] = 0.f;
    }
    __syncthreads();

    stageA_fullN<8, 6>(hbuf, tbuf, W1, b1, wave, lane);     __syncthreads();
    stageB_fullN(tbuf, hbuf, An, wave, lane, true);         __syncthreads();

    stageA_fullN<HH, HH>(hbuf, tbuf, W2, b2, wave, lane);   __syncthreads();
    stageB_fullN(tbuf, hbuf, An, wave, lane, true);         __syncthreads();

    stageA_fullN<HH, HH>(hbuf, tbuf, W3, b3, wave, lane);   __syncthreads();
    stageB_fullN(tbuf, hbuf, An, wave, lane, true);         __syncthreads();

    stageA_fullN<HH, HH>(hbuf, tbuf, W4, b4, wave, lane);   __syncthreads();
    stageB_fullN(tbuf, hbuf, An, wave, lane, true);         __syncthreads();

    layer5_stageA(hbuf, tbuf, W5, b5, wave, lane);          __syncthreads();
    layer5_stageB(tbuf, An, inp_b, out_b, wave, lane);
}

// ---------------------------------------------------------------------------
extern "C" void kernel_launch(void* const* d_in, const int* in_sizes, int n_in,
                              void* d_out, int out_size, void* d_ws, size_t ws_size,
                              hipStream_t stream) {
    const float* x      = (const float*)d_in[0];
    const float* inputs = (const float*)d_in[1];
    const float* adj    = (const float*)d_in[2];
    const float* W1 = (const float*)d_in[3];  const float* b1 = (const float*)d_in[4];
    const float* W2 = (const float*)d_in[5];  const float* b2 = (const float*)d_in[6];
    const float* W3 = (const float*)d_in[7];  const float* b3 = (const float*)d_in[8];
    const float* W4 = (const float*)d_in[9];  const float* b4 = (const float*)d_in[10];
    const float* W5 = (const float*)d_in[11]; const float* b5 = (const float*)d_in[12];
    float* out = (float*)d_out;
    float* An  = (float*)d_ws;                // 192*192*4 = 147456 B of scratch

    prep_An_kernel<<<1, 256, 0, stream>>>(adj, An);

    const size_t shmem = (size_t)2 * NN * HS * sizeof(float);  // ~198 KB of 320 KB LDS
    gcn_main_kernel<<<1024, 256, shmem, stream>>>(x, inputs, An,
                                                  W1, b1, W2, b2, W3, b3, W4, b4,
                                                  W5, b5, out);
}